// ImageMultiply_31129922962040
// MI455X (gfx1250) — compile-verified
//
#include <hip/hip_runtime.h>
#include <hip/hip_bf16.h>

// Problem constants from the reference: B=32, C=4, L=512, f32 in/out.
#define BB 32
#define CC 4
#define LL 512
#define ROWS_PER_BLK 64
#define BLK 256

typedef __attribute__((ext_vector_type(2))) float v2f;
typedef __attribute__((ext_vector_type(8))) float v8f;

// Full wave32 reduction. Uses V_WMMA_F32_16X16X4_F32 (the only f32 WMMA shape)
// as a one-shot cross-lane adder:
//   A (16x4): lanes 0-15 -> {K0=acc, K1=0}, lanes 16-31 -> {K2=acc, K3=0}
//   B (4x16): all ones  =>  D[m][n] = acc[m] + acc[m+16]
//   Sum of 8 D VGPRs per lane = half-wave sum; shfl_xor(16) completes it.
__device__ __forceinline__ float wave_reduce_sum(float acc) {
#if __has_builtin(__builtin_amdgcn_wmma_f32_16x16x4_f32)
  v2f a; a.x = acc;  a.y = 0.0f;
  v2f b; b.x = 1.0f; b.y = 1.0f;
  v8f c = {};
  v8f d = __builtin_amdgcn_wmma_f32_16x16x4_f32(
      /*neg_a=*/false, a, /*neg_b=*/false, b,
      /*c_mod=*/(short)0, c, /*reuse_a=*/false, /*reuse_b=*/false);
  float s = ((d[0] + d[1]) + (d[2] + d[3])) + ((d[4] + d[5]) + (d[6] + d[7]));
  s += __shfl_xor(s, 16, 32);
  return s;
#else
  float s = acc;
  s += __shfl_xor(s, 16, 32);
  s += __shfl_xor(s, 8, 32);
  s += __shfl_xor(s, 4, 32);
  s += __shfl_xor(s, 2, 32);
  s += __shfl_xor(s, 1, 32);
  return s;
#endif
}

__global__ void zero_out_kernel(float* out, int n) {
  int i = blockIdx.x * blockDim.x + threadIdx.x;
  if (i < n) out[i] = 0.0f;
}

// grid = (L/ROWS_PER_BLK, C, B), block = 256 threads (8 waves)
__global__ void shift_corr_kernel(const float* __restrict__ receptor,
                                  const float* __restrict__ ligand,
                                  const int* __restrict__ T,
                                  float* __restrict__ out) {
  const int b = blockIdx.z;
  const int c = blockIdx.y;
  const int row0 = blockIdx.x * ROWS_PER_BLK;

  const int dx = T[2 * b + 0];
  const int dy = T[2 * b + 1];

  const size_t plane = ((size_t)(b * CC + c)) * (size_t)LL * (size_t)LL;
  const float* __restrict__ rp = receptor + plane;
  const float* __restrict__ lp = ligand + plane;

  // Branch-free valid column range: y in [y_lo, y_hi) <=> 0 <= y-dy < L
  const int y_lo = dy > 0 ? dy : 0;
  const int y_hi = (LL + dy) < LL ? (LL + dy) : LL;

  float acc = 0.0f;

  for (int xr = 0; xr < ROWS_PER_BLK; ++xr) {
    const int x = row0 + xr;
    const int ix = x - dx;               // block-uniform: no lane divergence
    if (ix < 0 || ix >= LL) continue;
    const float* __restrict__ rrow = rp + (size_t)x * LL;
    const float* __restrict__ lrow = lp + (size_t)ix * LL;
    // Coalesced: lane t streams columns t, t+256; ligand row offset by dy.
    for (int y = y_lo + (int)threadIdx.x; y < y_hi; y += BLK) {
      acc = fmaf(rrow[y], lrow[y - dy], acc);
    }
  }

  // EXEC is full here (uniform control flow) -> WMMA legal.
  const float wsum = wave_reduce_sum(acc);

  __shared__ float smem[BLK / 32];
  const int wave = threadIdx.x >> 5;
  const int lane = threadIdx.x & 31;
  if (lane == 0) smem[wave] = wsum;
  __syncthreads();

  if (threadIdx.x == 0) {
    float t = 0.0f;
#pragma unroll
    for (int i = 0; i < BLK / 32; ++i) t += smem[i];
    atomicAdd(out + b, t);
  }
}

extern "C" void kernel_launch(void* const* d_in, const int* in_sizes, int n_in,
                              void* d_out, int out_size, void* d_ws, size_t ws_size,
                              hipStream_t stream) {
  const float* receptor = (const float*)d_in[0];
  const float* ligand   = (const float*)d_in[1];
  const int*   T        = (const int*)d_in[2];
  float* out = (float*)d_out;

  // d_out is poisoned by the harness and we accumulate atomically: zero first.
  zero_out_kernel<<<1, 32, 0, stream>>>(out, out_size);

  dim3 grid(LL / ROWS_PER_BLK, CC, BB);  // 8 x 4 x 32 = 1024 blocks
  shift_corr_kernel<<<grid, BLK, 0, stream>>>(receptor, ligand, T, out);
}